// RestoreNet_no_rotate_90228672954719
// MI455X (gfx1250) — compile-verified
//
#include <hip/hip_runtime.h>
#include <math.h>

// Problem constants (match reference setup_inputs)
constexpr int  cB = 8;
constexpr int  cN = 2048;
constexpr int  cM = 2048;
constexpr int  cD = 256;
constexpr int  K_SIM  = 16;
constexpr int  K_COOR = 4;
#define EPSV 1e-8f

typedef __attribute__((ext_vector_type(2))) float v2f;
typedef __attribute__((ext_vector_type(8))) float v8f;

// ---------------------------------------------------------------------------
// Kernel A: per-row inverse norms  inv[r] = 1 / max(||f[r]||, eps)
// one wave32 per row of D=256 floats
// ---------------------------------------------------------------------------
__global__ void rownorm_kernel(const float* __restrict__ f,
                               float* __restrict__ inv, int nrows) {
    int wave = (int)((blockIdx.x * blockDim.x + threadIdx.x) >> 5);
    int lane = threadIdx.x & 31;
    if (wave >= nrows) return;
    const float* row = f + (size_t)wave * cD;
    float s = 0.f;
    #pragma unroll
    for (int d = 0; d < cD / 32; ++d) {
        float v = row[lane + d * 32];
        s += v * v;
    }
    #pragma unroll
    for (int m = 16; m >= 1; m >>= 1) s += __shfl_xor(s, m, 32);
    if (lane == 0) inv[wave] = 1.0f / fmaxf(sqrtf(s), EPSV);
}

// ---------------------------------------------------------------------------
// Kernel B: coordinate similarity top-4 per (b, m) row.
// w2[m,n] = exp(-(|q|^2 + |p|^2 - 2 q.p)); keep top-4 (stable: smaller n wins ties)
// one wave32 per row; per-lane sorted-4 lists merged via LDS.
// ---------------------------------------------------------------------------
__global__ void coord_top4_kernel(const float* __restrict__ p,
                                  const float* __restrict__ q,
                                  float* __restrict__ t4v,
                                  int*   __restrict__ t4i) {
    __shared__ float sv[8][32 * K_COOR];
    __shared__ int   si[8][32 * K_COOR];
    int wv   = threadIdx.x >> 5;
    int lane = threadIdx.x & 31;
    int row  = (int)(blockIdx.x * 8 + wv);       // row = b*M + m
    int b    = row / cM;

    const float* qr = q + (size_t)row * 3;
    float qx = qr[0], qy = qr[1], qz = qr[2];
    float qq = qx * qx + qy * qy + qz * qz;

    float vals[K_COOR];
    int   idxs[K_COOR];
    #pragma unroll
    for (int t = 0; t < K_COOR; ++t) { vals[t] = -INFINITY; idxs[t] = 0x7fffffff; }

    for (int n = lane; n < cN; n += 32) {
        const float* pr = p + ((size_t)b * cN + n) * 3;
        float px = pr[0], py = pr[1], pz = pr[2];
        float pp = px * px + py * py + pz * pz;
        float d2 = qq + pp - 2.0f * (qx * px + qy * py + qz * pz);
        float v  = expf(-d2);
        if (v > vals[K_COOR - 1]) {              // strict: earlier (smaller n) wins ties
            int t = K_COOR - 1;
            while (t > 0 && v > vals[t - 1]) {
                vals[t] = vals[t - 1]; idxs[t] = idxs[t - 1]; --t;
            }
            vals[t] = v; idxs[t] = n;
        }
    }
    #pragma unroll
    for (int t = 0; t < K_COOR; ++t) {
        sv[wv][lane * K_COOR + t] = vals[t];
        si[wv][lane * K_COOR + t] = idxs[t];
    }
    __syncthreads();
    if (lane == 0) {
        int heads[32];
        #pragma unroll
        for (int l = 0; l < 32; ++l) heads[l] = 0;
        for (int r = 0; r < K_COOR; ++r) {
            float best = -INFINITY; int bi = 0x7fffffff; int bl = 0;
            for (int l = 0; l < 32; ++l) {
                if (heads[l] < K_COOR) {
                    float v = sv[wv][l * K_COOR + heads[l]];
                    int   i = si[wv][l * K_COOR + heads[l]];
                    if (v > best || (v == best && i < bi)) { best = v; bi = i; bl = l; }
                }
            }
            heads[bl]++;
            t4v[(size_t)row * K_COOR + r] = best;
            t4i[(size_t)row * K_COOR + r] = bi;
        }
    }
}

// ---------------------------------------------------------------------------
// Kernel C: w = exp((f2 . f1^T) * inv2[m] * inv1[n])
// One wave32 per 32x64 output block: 2x4 register-blocked 16x16 WMMA tiles
// (64 accumulator VGPRs -> no spills, 2+ waves/SIMD for latency hiding).
// Per k-step: 2 A frags + 4 B frags -> 8 V_WMMA_F32_16X16X4_F32 ops.
// A frag (16x4 f32): lane l -> M=l%16, {K = k+2*(l/16), k+2*(l/16)+1}
// B frag (4x16 f32): lane l -> N=l%16, same K pair (data = f1 row slices)
// C/D (16x16 f32): VGPR r, lane l -> M = r + 8*(l/16), N = l%16
// Row offsets (i*16*cD floats) fold into the 24-bit instruction offset.
// ---------------------------------------------------------------------------
constexpr int WMT = 2;   // 16-row tiles per wave (M)
constexpr int WNT = 4;   // 16-col tiles per wave (N)

__global__ void __launch_bounds__(256)
gemm_w_kernel(const float* __restrict__ f1,
              const float* __restrict__ f2,
              const float* __restrict__ inv1,
              const float* __restrict__ inv2,
              float* __restrict__ w_out) {
    constexpr int TM = cM / (16 * WMT);          // 64 blocks
    constexpr int TN = cN / (16 * WNT);          // 32 blocks
    int gw   = (int)((blockIdx.x * blockDim.x + threadIdx.x) >> 5);
    int lane = threadIdx.x & 31;
    int b    = gw / (TM * TN);
    int rem  = gw % (TM * TN);
    int mt   = rem / TN;
    int nt   = rem % TN;
    int half = lane >> 4;                        // 0 or 1
    int l16  = lane & 15;
    int m0 = mt * 16 * WMT, n0 = nt * 16 * WNT;

    const float* Abase = f2 + ((size_t)b * cM + m0 + l16) * cD + 2 * half;
    const float* Bbase = f1 + ((size_t)b * cN + n0 + l16) * cD + 2 * half;

    v8f acc[WMT][WNT];
    #pragma unroll
    for (int i = 0; i < WMT; ++i)
        #pragma unroll
        for (int j = 0; j < WNT; ++j)
            acc[i][j] = (v8f){};

    #pragma unroll 2
    for (int k = 0; k < cD; k += 4) {
        v2f a[WMT], bb[WNT];
        #pragma unroll
        for (int i = 0; i < WMT; ++i)
            a[i] = *(const v2f*)(Abase + (size_t)i * 16 * cD + k);
        #pragma unroll
        for (int j = 0; j < WNT; ++j)
            bb[j] = *(const v2f*)(Bbase + (size_t)j * 16 * cD + k);
        #pragma unroll
        for (int i = 0; i < WMT; ++i)
            #pragma unroll
            for (int j = 0; j < WNT; ++j)
                acc[i][j] = __builtin_amdgcn_wmma_f32_16x16x4_f32(
                    false, a[i], false, bb[j], (short)0, acc[i][j], false, false);
    }

    // epilogue: scale by inverse norms, exp, store
    float invn[WNT];
    #pragma unroll
    for (int j = 0; j < WNT; ++j)
        invn[j] = inv1[(size_t)b * cN + n0 + j * 16 + l16];

    #pragma unroll
    for (int i = 0; i < WMT; ++i) {
        #pragma unroll
        for (int r = 0; r < 8; ++r) {
            int    mm   = m0 + i * 16 + r + 8 * half;
            size_t rowi = (size_t)b * cM + mm;
            float  i2   = inv2[rowi];
            float* wrow = w_out + rowi * cN + n0 + l16;
            #pragma unroll
            for (int j = 0; j < WNT; ++j)
                wrow[j * 16] = expf(acc[i][j][r] * i2 * invn[j]);
        }
    }
}

// ---------------------------------------------------------------------------
// Kernel C2: scatter-add the sparse coordinate weights:
// w[row, t4i[row][t]] += t4v[row][t]   (top-k indices are distinct per row)
// ---------------------------------------------------------------------------
__global__ void scatter_w2_kernel(const float* __restrict__ t4v,
                                  const int*   __restrict__ t4i,
                                  float* __restrict__ w) {
    int tid = (int)(blockIdx.x * blockDim.x + threadIdx.x);  // over rows*K_COOR
    if (tid >= cB * cM * K_COOR) return;
    int row = tid >> 2;
    int idx = t4i[tid];
    w[(size_t)row * cN + idx] += t4v[tid];
}

// ---------------------------------------------------------------------------
// Kernel D: top-16 per row of w (stable: smaller index wins ties, like
// jax.lax.top_k). One wave32 per row; per-lane sorted-16 lists, LDS 32-way merge.
// Writes idx as float into d_out region and as int into workspace.
// ---------------------------------------------------------------------------
__global__ void top16_kernel(const float* __restrict__ w,
                             float* __restrict__ out_idxf,
                             int*   __restrict__ top16) {
    __shared__ float sv[8][32 * K_SIM];
    __shared__ int   si[8][32 * K_SIM];
    int wv   = threadIdx.x >> 5;
    int lane = threadIdx.x & 31;
    int row  = (int)(blockIdx.x * 8 + wv);       // row = b*M + m
    const float* wr = w + (size_t)row * cN;

    float vals[K_SIM];
    int   idxs[K_SIM];
    #pragma unroll
    for (int t = 0; t < K_SIM; ++t) { vals[t] = -INFINITY; idxs[t] = 0x7fffffff; }

    for (int n = lane; n < cN; n += 32) {
        float v = wr[n];
        if (v > vals[K_SIM - 1]) {
            int t = K_SIM - 1;
            while (t > 0 && v > vals[t - 1]) {
                vals[t] = vals[t - 1]; idxs[t] = idxs[t - 1]; --t;
            }
            vals[t] = v; idxs[t] = n;
        }
    }
    #pragma unroll
    for (int t = 0; t < K_SIM; ++t) {
        sv[wv][lane * K_SIM + t] = vals[t];
        si[wv][lane * K_SIM + t] = idxs[t];
    }
    __syncthreads();
    if (lane == 0) {
        int heads[32];
        #pragma unroll
        for (int l = 0; l < 32; ++l) heads[l] = 0;
        for (int r = 0; r < K_SIM; ++r) {
            float best = -INFINITY; int bi = 0x7fffffff; int bl = 0;
            for (int l = 0; l < 32; ++l) {
                if (heads[l] < K_SIM) {
                    float v = sv[wv][l * K_SIM + heads[l]];
                    int   i = si[wv][l * K_SIM + heads[l]];
                    if (v > best || (v == best && i < bi)) { best = v; bi = i; bl = l; }
                }
            }
            heads[bl]++;
            out_idxf[(size_t)row * K_SIM + r] = (float)bi;
            top16  [(size_t)row * K_SIM + r] = bi;
        }
    }
}

// ---------------------------------------------------------------------------
// Kernel E: gather 16 f1 rows per (b,m) and fused mean/max pool -> f [B,M,2D]
// one block (256 threads = one d each) per output row.
// ---------------------------------------------------------------------------
__global__ void pool_kernel(const float* __restrict__ f1,
                            const int* __restrict__ top16,
                            float* __restrict__ out_f) {
    __shared__ int sidx[K_SIM];
    int bm = (int)blockIdx.x;                    // b*M + m
    int d  = threadIdx.x;                        // 0..255
    int b  = bm / cM;
    if (threadIdx.x < K_SIM)
        sidx[threadIdx.x] = top16[(size_t)bm * K_SIM + threadIdx.x];
    __syncthreads();
    float mx = -INFINITY, sum = 0.f;
    #pragma unroll
    for (int k = 0; k < K_SIM; ++k) {
        float v = f1[((size_t)b * cN + sidx[k]) * cD + d];
        mx  = fmaxf(mx, v);
        sum += v;
    }
    out_f[(size_t)bm * (2 * cD) + d]      = sum * (1.0f / (float)K_SIM);
    out_f[(size_t)bm * (2 * cD) + cD + d] = mx;
}

// ---------------------------------------------------------------------------
extern "C" void kernel_launch(void* const* d_in, const int* in_sizes, int n_in,
                              void* d_out, int out_size, void* d_ws, size_t ws_size,
                              hipStream_t stream) {
    const float* f1 = (const float*)d_in[0];     // [B,N,D]
    const float* f2 = (const float*)d_in[1];     // [B,M,D]
    const float* p  = (const float*)d_in[2];     // [B,N,3]
    const float* q  = (const float*)d_in[3];     // [B,M,3]

    // outputs packed flat in return order: f, idx, w
    float* out_f    = (float*)d_out;                                 // B*M*2D
    float* out_idxf = out_f + (size_t)cB * cM * 2 * cD;              // B*M*16
    float* out_w    = out_idxf + (size_t)cB * cM * K_SIM;            // B*M*N

    // workspace carve-out
    char*  ws   = (char*)d_ws;
    float* inv1 = (float*)ws;  ws += sizeof(float) * (size_t)cB * cN;
    float* inv2 = (float*)ws;  ws += sizeof(float) * (size_t)cB * cM;
    float* t4v  = (float*)ws;  ws += sizeof(float) * (size_t)cB * cM * K_COOR;
    int*   t4i  = (int*)ws;    ws += sizeof(int)   * (size_t)cB * cM * K_COOR;
    int*   t16  = (int*)ws;    ws += sizeof(int)   * (size_t)cB * cM * K_SIM;
    (void)in_sizes; (void)n_in; (void)out_size; (void)ws_size;

    const int rows1 = cB * cN;                   // 16384
    const int rows2 = cB * cM;                   // 16384

    rownorm_kernel<<<rows1 / 8, 256, 0, stream>>>(f1, inv1, rows1);
    rownorm_kernel<<<rows2 / 8, 256, 0, stream>>>(f2, inv2, rows2);
    coord_top4_kernel<<<rows2 / 8, 256, 0, stream>>>(p, q, t4v, t4i);

    const int total_waves = cB * (cM / (16 * WMT)) * (cN / (16 * WNT)); // 16384
    gemm_w_kernel<<<total_waves / 8, 256, 0, stream>>>(f1, f2, inv1, inv2, out_w);

    scatter_w2_kernel<<<(rows2 * K_COOR) / 256, 256, 0, stream>>>(t4v, t4i, out_w);

    top16_kernel<<<rows2 / 8, 256, 0, stream>>>(out_w, out_idxf, t16);
    pool_kernel<<<rows2, 256, 0, stream>>>(f1, t16, out_f);
}